// SO3Convolution_44581760533231
// MI455X (gfx1250) — compile-verified
//
#include <hip/hip_runtime.h>
#include <math.h>

// ---------------------------------------------------------------------------
// SO(3) convolution for MI455X (gfx1250, wave32, WMMA).
// Pipeline:
//   K1  partial 2-D DFT (only |m|<=7 freqs) per (z,f,b) slice  -> XF   [WMMA f32]
//   K2a Wigner-forward contraction over beta                    -> XH
//   K2b kernel x Fourier-table -> conj(yhat), laid out as B mats-> Y
//   K3  per-degree batched complex GEMM  zl = xhat * conj(y)    -> ZL  [WMMA f32]
//   K4  inverse-Wigner scatter into centered 15x15 out-spectrum -> FF (reuses XF)
//   K5a/K5b inverse 16x16 DFT (real part) + bias                -> out
// ---------------------------------------------------------------------------

typedef __attribute__((ext_vector_type(2))) float v2f;
typedef __attribute__((ext_vector_type(8))) float v8f;

#define Z     32   // batch
#define FIN   16
#define GOUT  32
#define NB    32   // 2*B_IN
#define NBO   16   // 2*B_OUT
#define PF    15   // centered freqs m=-7..7
#define NSPEC 680
#define NGRID 24

__device__ __forceinline__ int s0_of(int l) { return l * (4 * l * l - 1) / 3; }

__device__ __forceinline__ v8f wmma4(v2f a, v2f b, v8f c) {
  // V_WMMA_F32_16X16X4_F32 : C += A(16x4) * B(4x16), all fp32
  return __builtin_amdgcn_wmma_f32_16x16x4_f32(false, a, false, b, (short)0, c,
                                               false, false);
}

// e^{-2*pi*i*mu/32}, exact integer range reduction
__device__ __forceinline__ void cis_n32(int mu, float& c, float& s) {
  int t = mu & 31;  // two's-complement AND == non-negative mod 32
  float ang = -0.19634954084936207f * (float)t;  // 2*pi/32
  c = __cosf(ang);
  s = __sinf(ang);
}

// ---------------- K1: partial forward DFT2, one wave per (z,f,b) slice ------
__global__ void k1_dft2_fwd(const float* __restrict__ x,
                            float* __restrict__ xf_re,
                            float* __restrict__ xf_im) {
  __shared__ float xs[NB * NB];
  __shared__ float tre[16 * NB];
  __shared__ float tim[16 * NB];
  const int slice = blockIdx.x;           // (z*16+f)*32+b
  const int lane  = threadIdx.x;          // 32 threads = 1 wave
  const float* xg = x + (size_t)slice * NB * NB;
  for (int i = 0; i < NB; ++i) xs[i * NB + lane] = xg[i * NB + lane];
  __syncthreads();

  const int mrow  = lane & 15;            // A-matrix row (output freq index r)
  const int khalf = (lane >> 4) << 1;     // K split per ISA A layout
  const int m     = mrow - 7;             // centered freq (row 15 is discarded pad)

  // stage 1: T(16x32) = W(16x32) * xs(32x32), W[r,u]=e^{-2pi i (r-7) u/32}
  v8f aR0 = {}, aI0 = {}, aR1 = {}, aI1 = {};
  for (int k0 = 0; k0 < NB; k0 += 4) {
    const int u0 = k0 + khalf;
    v2f ar, ai, b0, b1;
    float c0, s0, c1, s1;
    cis_n32(m * u0, c0, s0);
    cis_n32(m * (u0 + 1), c1, s1);
    ar.x = c0; ar.y = c1;
    ai.x = s0; ai.y = s1;
    b0.x = xs[u0 * NB + (lane & 15)];
    b0.y = xs[(u0 + 1) * NB + (lane & 15)];
    b1.x = xs[u0 * NB + 16 + (lane & 15)];
    b1.y = xs[(u0 + 1) * NB + 16 + (lane & 15)];
    aR0 = wmma4(ar, b0, aR0);  aI0 = wmma4(ai, b0, aI0);
    aR1 = wmma4(ar, b1, aR1);  aI1 = wmma4(ai, b1, aI1);
  }
#pragma unroll
  for (int i = 0; i < 8; ++i) {
    int row = i + ((lane >> 4) << 3);
    int col = lane & 15;
    tre[row * NB + col]      = aR0[i];
    tim[row * NB + col]      = aI0[i];
    tre[row * NB + 16 + col] = aR1[i];
    tim[row * NB + 16 + col] = aI1[i];
  }
  __syncthreads();

  // stage 2: XF(16x16) = T(16x32) * D(32x16), D[v,s]=e^{-2pi i (s-7) v/32}
  const int scol = lane & 15;
  v8f accR = {}, accI = {};
  for (int k0 = 0; k0 < NB; k0 += 4) {
    const int v0 = k0 + khalf;
    v2f tr, ti, nti, dr, di;
    tr.x = tre[(lane & 15) * NB + v0]; tr.y = tre[(lane & 15) * NB + v0 + 1];
    ti.x = tim[(lane & 15) * NB + v0]; ti.y = tim[(lane & 15) * NB + v0 + 1];
    nti.x = -ti.x; nti.y = -ti.y;
    float c0, s0, c1, s1;
    cis_n32((scol - 7) * v0, c0, s0);
    cis_n32((scol - 7) * (v0 + 1), c1, s1);
    dr.x = c0; dr.y = c1;
    di.x = s0; di.y = s1;
    accR = wmma4(tr,  dr, accR);
    accR = wmma4(nti, di, accR);
    accI = wmma4(tr,  di, accI);
    accI = wmma4(ti,  dr, accI);
  }
#pragma unroll
  for (int i = 0; i < 8; ++i) {
    int r = i + ((lane >> 4) << 3);
    if (r < PF && scol < PF) {
      size_t o = (size_t)slice * 225 + r * 15 + scol;
      xf_re[o] = accR[i];
      xf_im[o] = accI[i];
    }
  }
}

// ---------------- K2a: xhat[l][z,m,k,f] = sum_b wf[b,s] * XF[z,f,b,..] ------
__global__ void k2a_wigfwd(const float* __restrict__ wf,
                           const float* __restrict__ xf_re,
                           const float* __restrict__ xf_im,
                           float* __restrict__ xh_re,
                           float* __restrict__ xh_im) {
  const int s = blockIdx.x;
  int l = 0;
  while (s >= s0_of(l + 1)) l++;
  const int dim = 2 * l + 1;
  const int r = s - s0_of(l);
  const int m = r / dim, k = r % dim;
  const int t = threadIdx.x;              // 512 = f*32+z
  const int f = t >> 5, z = t & 31;
  const int Kl = dim * 16;
  float ar = 0.f, ai = 0.f;
  for (int b = 0; b < NB; ++b) {
    float w = wf[b * NSPEC + s];
    size_t o = ((size_t)((z * FIN + f) * NB + b)) * 225 +
               (7 - l + m) * 15 + (7 - l + k);
    ar += w * xf_re[o];
    ai += w * xf_im[o];
  }
  size_t oo = (size_t)8192 * l * l + (size_t)(z * 16 + m) * Kl + k * 16 + f;
  xh_re[oo] = ar;
  xh_im[oo] = ai;
}

// ---------------- K2b: B[(k,f),(n,g)] = conj(SCALING * kernel x Fc) ---------
__global__ void k2b_yhat(const float* __restrict__ kern,
                         const float* __restrict__ ftre,
                         const float* __restrict__ ftim,
                         float* __restrict__ y_re,
                         float* __restrict__ y_im) {
  const int s = blockIdx.x;
  int l = 0;
  while (s >= s0_of(l + 1)) l++;
  const int dim = 2 * l + 1;
  const int r = s - s0_of(l);
  const int n = r / dim, k = r % dim;
  const int t = threadIdx.x;              // 512 = f*32+g
  const int f = t >> 5, g = t & 31;
  float accr = 0.f, acci = 0.f;
  for (int p = 0; p < NGRID; ++p) {
    float kv = kern[(f * GOUT + g) * NGRID + p];
    accr += kv * ftre[p * NSPEC + s];
    acci += kv * ftim[p * NSPEC + s];
  }
  const int Nl = dim * 32;
  const float SC = 0.14433756729740644f;  // 1/sqrt(48)
  size_t o = (size_t)512 * s0_of(l) + (size_t)(k * 16 + f) * Nl + n * 32 + g;
  y_re[o] = SC * accr;
  y_im[o] = -SC * acci;                   // conj
}

// ---------------- K3: zl = xhat * conj(y), complex WMMA GEMM ----------------
__global__ void k3_zl(const float* __restrict__ xh_re,
                      const float* __restrict__ xh_im,
                      const float* __restrict__ y_re,
                      const float* __restrict__ y_im,
                      float* __restrict__ zl_re,
                      float* __restrict__ zl_im) {
  const int lane = threadIdx.x;           // one wave per block
  const int z = blockIdx.x & 31;
  const int t128 = blockIdx.x >> 5;       // 0..127: (l, n-tile)
  int l = 0;
  while (2 * (l + 1) * (l + 1) <= t128) l++;
  const int nt = t128 - 2 * l * l;        // 0 .. 2*(2l+1)-1
  const int dim = 2 * l + 1;
  const int Kl = dim * 16, Nl = dim * 32;
  const float* axr = xh_re + (size_t)8192 * l * l + (size_t)(z * 16) * Kl;
  const float* axi = xh_im + (size_t)8192 * l * l + (size_t)(z * 16) * Kl;
  const float* byr = y_re + (size_t)512 * s0_of(l);
  const float* byi = y_im + (size_t)512 * s0_of(l);
  const int mrow = lane & 15;
  const int khalf = (lane >> 4) << 1;
  const int col = nt * 16 + (lane & 15);
  v8f accR = {}, accI = {};
  for (int k0 = 0; k0 < Kl; k0 += 4) {
    const int kk = k0 + khalf;
    v2f ar, ai, nai, br, bi;
    ar.x = axr[mrow * Kl + kk];  ar.y = axr[mrow * Kl + kk + 1];
    ai.x = axi[mrow * Kl + kk];  ai.y = axi[mrow * Kl + kk + 1];
    nai.x = -ai.x; nai.y = -ai.y;
    br.x = byr[kk * Nl + col];   br.y = byr[(kk + 1) * Nl + col];
    bi.x = byi[kk * Nl + col];   bi.y = byi[(kk + 1) * Nl + col];
    accR = wmma4(ar,  br, accR);
    accR = wmma4(nai, bi, accR);   // re -= Ai*Bi
    accI = wmma4(ar,  bi, accI);
    accI = wmma4(ai,  br, accI);
  }
  float* zr = zl_re + (size_t)16384 * l * l + (size_t)(z * 16) * Nl;
  float* zi = zl_im + (size_t)16384 * l * l + (size_t)(z * 16) * Nl;
#pragma unroll
  for (int i = 0; i < 8; ++i) {
    int row = i + ((lane >> 4) << 3);
    zr[row * Nl + col] = accR[i];
    zi[row * Nl + col] = accI[i];
  }
}

// ---------------- K4: inverse-Wigner scatter into centered 15x15 spectrum ---
__global__ void k4_wiginv(const float* __restrict__ wi,
                          const float* __restrict__ zl_re,
                          const float* __restrict__ zl_im,
                          float* __restrict__ ff_re,
                          float* __restrict__ ff_im) {
  const int tid = blockIdx.x * blockDim.x + threadIdx.x;  // 16384*225
  const int nc = tid % 15;
  const int mc = (tid / 15) % 15;
  const int slice = tid / 225;            // (z*32+g)*16+b
  const int b = slice & 15;
  const int g = (slice >> 4) & 31;
  const int z = slice >> 9;
  const int m0 = mc - 7, n0 = nc - 7;
  const int am = m0 < 0 ? -m0 : m0, an = n0 < 0 ? -n0 : n0;
  const int lmin = am > an ? am : an;
  float ar = 0.f, ai = 0.f;
  for (int l = lmin; l < 8; ++l) {
    int dim = 2 * l + 1, Nl = dim * 32;
    int mi = m0 + l, ni = n0 + l;
    float w = wi[b * NSPEC + s0_of(l) + mi * dim + ni];
    size_t o = (size_t)16384 * l * l + (size_t)(z * 16 + mi) * Nl + ni * 32 + g;
    ar += w * zl_re[o];
    ai += w * zl_im[o];
  }
  ff_re[(size_t)slice * 225 + mc * 15 + nc] = ar;
  ff_im[(size_t)slice * 225 + mc * 15 + nc] = ai;
}

// ---------------- K5a/K5b: inverse 16x16 DFT (real part), + bias ------------
__global__ void k5a_idft_rows(const float* __restrict__ ff_re,
                              const float* __restrict__ ff_im,
                              float* __restrict__ sf_re,
                              float* __restrict__ sf_im) {
  const int tid = blockIdx.x * blockDim.x + threadIdx.x;  // 16384*240
  const int nc = tid % 15;
  const int uu = (tid / 15) & 15;
  const int slice = tid / 240;
  float ar = 0.f, ai = 0.f;
  for (int mc = 0; mc < 15; ++mc) {
    int t = ((mc - 7) * uu) & 15;
    float ang = 0.39269908169872414f * (float)t;  // +2*pi/16
    float c = __cosf(ang), s = __sinf(ang);
    float fr = ff_re[(size_t)slice * 225 + mc * 15 + nc];
    float fi = ff_im[(size_t)slice * 225 + mc * 15 + nc];
    ar += fr * c - fi * s;
    ai += fr * s + fi * c;
  }
  sf_re[(size_t)slice * 240 + uu * 15 + nc] = ar;
  sf_im[(size_t)slice * 240 + uu * 15 + nc] = ai;
}

__global__ void k5b_idft_cols(const float* __restrict__ sf_re,
                              const float* __restrict__ sf_im,
                              const float* __restrict__ bias,
                              float* __restrict__ out) {
  const int tid = blockIdx.x * blockDim.x + threadIdx.x;  // 16384*256 == out
  const int vv = tid & 15;
  const int uu = (tid >> 4) & 15;
  const int slice = tid >> 8;
  const int g = (slice >> 4) & 31;
  float acc = 0.f;
  for (int nc = 0; nc < 15; ++nc) {
    int t = ((nc - 7) * vv) & 15;
    float ang = 0.39269908169872414f * (float)t;
    acc += sf_re[(size_t)slice * 240 + uu * 15 + nc] * __cosf(ang) -
           sf_im[(size_t)slice * 240 + uu * 15 + nc] * __sinf(ang);
  }
  out[tid] = acc + bias[g];
}

// ---------------------------------------------------------------------------
extern "C" void kernel_launch(void* const* d_in, const int* in_sizes, int n_in,
                              void* d_out, int out_size, void* d_ws,
                              size_t ws_size, hipStream_t stream) {
  const float* x    = (const float*)d_in[0];
  const float* kern = (const float*)d_in[1];
  const float* bias = (const float*)d_in[2];
  const float* wf   = (const float*)d_in[3];
  const float* wi   = (const float*)d_in[4];
  const float* ftre = (const float*)d_in[5];
  const float* ftim = (const float*)d_in[6];

  float* ws = (float*)d_ws;
  constexpr size_t NXF   = (size_t)16384 * 225;  // centered 15x15 spectra
  constexpr size_t XH_SZ = 524288;               // sum_l 32*16*(2l+1)*16
  constexpr size_t Y_SZ  = 348160;               // 512 * 680
  constexpr size_t ZL_SZ = 1048576;              // sum_l 32*16*(2l+1)*32
  constexpr size_t SF_SZ = (size_t)16384 * 240;

  float* xf_re = ws;                float* xf_im = xf_re + NXF;  // reused as FF
  float* xh_re = xf_im + NXF;       float* xh_im = xh_re + XH_SZ;
  float* y_re  = xh_im + XH_SZ;     float* y_im  = y_re + Y_SZ;
  float* zl_re = y_im + Y_SZ;       float* zl_im = zl_re + ZL_SZ;
  float* sf_re = zl_im + ZL_SZ;     float* sf_im = sf_re + SF_SZ;

  // zero padded rows of xhat (m >= 2l+1) so WMMA A-operand pad rows are clean
  hipMemsetAsync(xh_re, 0, 2 * XH_SZ * sizeof(float), stream);

  k1_dft2_fwd<<<16384, 32, 0, stream>>>(x, xf_re, xf_im);
  k2a_wigfwd<<<680, 512, 0, stream>>>(wf, xf_re, xf_im, xh_re, xh_im);
  k2b_yhat<<<680, 512, 0, stream>>>(kern, ftre, ftim, y_re, y_im);
  k3_zl<<<4096, 32, 0, stream>>>(xh_re, xh_im, y_re, y_im, zl_re, zl_im);
  k4_wiginv<<<14400, 256, 0, stream>>>(wi, zl_re, zl_im, xf_re, xf_im);
  k5a_idft_rows<<<15360, 256, 0, stream>>>(xf_re, xf_im, sf_re, sf_im);
  k5b_idft_cols<<<16384, 256, 0, stream>>>(sf_re, sf_im, bias, (float*)d_out);
}